// CoNCELoss_4913442587114
// MI455X (gfx1250) — compile-verified
//
#include <hip/hip_runtime.h>
#include <hip/hip_bf16.h>
#include <math.h>

typedef __attribute__((ext_vector_type(2))) float v2f;
typedef __attribute__((ext_vector_type(8))) float v8f;

#define BB   8
#define NP   1024
#define DIM  256
#define TAU  0.07f

// ---------------------------------------------------------------------------
// zero-init small vectors (workspace is poisoned by the harness)
// ---------------------------------------------------------------------------
__global__ void zero_f32(float* __restrict__ p, int n) {
    int i = blockIdx.x * blockDim.x + threadIdx.x;
    if (i < n) p[i] = 0.0f;
}

// ---------------------------------------------------------------------------
// S = Q · K^T  per batch, exact f32 via V_WMMA_F32_16X16X4_F32.
// One wave32 per 16x16 output tile; also writes the transposed copy ST
// so every later pass is unit-stride (both live in the 192MB L2).
// ---------------------------------------------------------------------------
__global__ __launch_bounds__(256) void gemm_qkT(const float* __restrict__ fq,
                                                const float* __restrict__ fk,
                                                float* __restrict__ S,
                                                float* __restrict__ ST) {
    const int wave = threadIdx.x >> 5;
    const int lane = threadIdx.x & 31;
    const int tile = blockIdx.x * 8 + wave;     // 8 * 64 * 64 = 32768 tiles
    const int b    = tile >> 12;                // 4096 tiles per batch
    const int tm   = (tile >> 6) & 63;
    const int tn   = tile & 63;
    const int half = lane >> 4;                 // 0: lanes 0-15, 1: lanes 16-31
    const int l16  = lane & 15;

    // A: 16x4 f32 tile of Q.  lane<16 holds K=0,1 ; lane>=16 holds K=2,3.
    // B: 4x16 f32 tile of K^T: B[k][n] = fk[n][k] -> identical load pattern.
    const float* Q = fq + ((size_t)b * NP + (size_t)(tm * 16 + l16)) * DIM;
    const float* K = fk + ((size_t)b * NP + (size_t)(tn * 16 + l16)) * DIM;

    v8f acc = {};
    #pragma unroll 4
    for (int kb = 0; kb < DIM; kb += 4) {
        v2f a  = *(const v2f*)(Q + kb + 2 * half);
        v2f bv = *(const v2f*)(K + kb + 2 * half);
        acc = __builtin_amdgcn_wmma_f32_16x16x4_f32(
            /*neg_a=*/false, a, /*neg_b=*/false, bv,
            /*c_mod=*/(short)0, acc, /*reuse_a=*/false, /*reuse_b=*/false);
    }

    // C/D layout: VGPR r, lanes 0-15 -> M = r, lanes 16-31 -> M = r + 8; N = l16
    const size_t base = (size_t)b * NP * NP;
    const int n = tn * 16 + l16;
    #pragma unroll
    for (int r = 0; r < 8; ++r) {
        int m = tm * 16 + r + 8 * half;
        float val = acc[r];
        S [base + (size_t)m * NP + n] = val;    // coalesced
        ST[base + (size_t)n * NP + m] = val;    // one-time scatter
    }
}

// ---------------------------------------------------------------------------
// Sinkhorn half-step (eps = 1, u-terms cancel inside the LSE):
//   vout[row] = 1 - log(NP) - LSE_n( Mrows[row, n] + vin[b, n] )
// One wave32 per row, online (max,sum) logsumexp + shuffle reduction.
// Called with (S, v) -> u  and (ST, u) -> v.
// ---------------------------------------------------------------------------
__global__ __launch_bounds__(256) void lse_update(const float* __restrict__ Mrows,
                                                  const float* __restrict__ vin,
                                                  float* __restrict__ vout) {
    const int row  = blockIdx.x * 8 + (threadIdx.x >> 5);   // 8192 rows
    const int lane = threadIdx.x & 31;
    const float* r  = Mrows + (size_t)row * NP;
    const float* vv = vin + (row & ~(NP - 1));              // batch base b*NP

    float mx = -INFINITY, sm = 0.0f;
    for (int n = lane; n < NP; n += 32) {
        float x  = r[n] + vv[n];
        float nm = fmaxf(mx, x);
        sm = sm * __expf(mx - nm) + __expf(x - nm);
        mx = nm;
    }
    #pragma unroll
    for (int off = 16; off; off >>= 1) {
        float mo = __shfl_xor(mx, off);
        float so = __shfl_xor(sm, off);
        float nm = fmaxf(mx, mo);
        sm = sm * __expf(mx - nm) + so * __expf(mo - nm);
        mx = nm;
    }
    if (lane == 0)
        vout[row] = 1.0f - 6.93147180559945f /*log 1024*/ - (mx + logf(sm));
}

// ---------------------------------------------------------------------------
// Final loss per row r = (b, m):
//   log T[b,n,m] = ST[b,m,n] + u[b,n] + v[b,m] - 1          (row of ST!)
//   f_max        = exp(max_n log T) + 1e-8
//   l_pos        = S[b,m,m] + 0.07*log(f_max)               (if i==4)
//   l_neg[n]     = S[b,m,n] + 0.07*log(exp(logT)+1e-8), diag -> -10
//   loss         = LSE([l_pos, l_neg]/tau) - l_pos/tau
// ---------------------------------------------------------------------------
__global__ __launch_bounds__(256) void loss_kernel(const float* __restrict__ S,
                                                   const float* __restrict__ ST,
                                                   const float* __restrict__ u,
                                                   const float* __restrict__ v,
                                                   const int* __restrict__ ip,
                                                   float* __restrict__ out) {
    const int row  = blockIdx.x * 8 + (threadIdx.x >> 5);   // 8192 rows
    const int lane = threadIdx.x & 31;
    const int b    = row >> 10;
    const int m    = row & (NP - 1);
    const float* srow  = S  + (size_t)row * NP;
    const float* strow = ST + (size_t)row * NP;
    const float* ub    = u + b * NP;
    const float  vm    = v[row];
    const bool   add   = (*ip == 4);
    const float  inv_tau = 1.0f / TAU;

    float gmax = -INFINITY;                 // running max of log T[b,:,m]
    float lmax = -INFINITY, lsum = 0.0f;    // online LSE of l_neg/tau

    for (int n = lane; n < NP; n += 32) {
        float s  = srow[n];
        float g  = strow[n] + ub[n] + vm - 1.0f;    // log T[b,n,m]
        gmax = fmaxf(gmax, g);
        float ln;
        if (n == m) {
            ln = -10.0f;
        } else {
            ln = s;
            if (add) ln += TAU * logf(__expf(g) + 1e-8f);
        }
        float o  = ln * inv_tau;
        float nm = fmaxf(lmax, o);
        lsum = lsum * __expf(lmax - nm) + __expf(o - nm);
        lmax = nm;
    }
    #pragma unroll
    for (int off = 16; off; off >>= 1) {
        gmax = fmaxf(gmax, __shfl_xor(gmax, off));
        float mo = __shfl_xor(lmax, off);
        float so = __shfl_xor(lsum, off);
        float nm = fmaxf(lmax, mo);
        lsum = lsum * __expf(lmax - nm) + so * __expf(mo - nm);
        lmax = nm;
    }
    if (lane == 0) {
        float fmaxv = __expf(gmax) + 1e-8f;             // exp is monotone
        float lpos  = srow[m] + (add ? TAU * logf(fmaxv) : 0.0f);
        float o0    = lpos * inv_tau;
        float Mv    = fmaxf(lmax, o0);
        float total = Mv + logf(lsum * __expf(lmax - Mv) + __expf(o0 - Mv));
        out[row]    = total - o0;
    }
}

// ---------------------------------------------------------------------------
extern "C" void kernel_launch(void* const* d_in, const int* in_sizes, int n_in,
                              void* d_out, int out_size, void* d_ws, size_t ws_size,
                              hipStream_t stream) {
    const float* fq = (const float*)d_in[0];   // [8192, 256]
    const float* fk = (const float*)d_in[1];   // [8192, 256]
    const int*   ip = (const int*)d_in[2];     // scalar i

    float* S  = (float*)d_ws;                  // [8,1024,1024]  32 MB
    float* ST = S  + (size_t)BB * NP * NP;     // transposed     32 MB
    float* u  = ST + (size_t)BB * NP * NP;     // [8,1024]
    float* v  = u  + BB * NP;                  // [8,1024]
    float* out = (float*)d_out;                // [8192]

    zero_f32<<<(2 * BB * NP + 255) / 256, 256, 0, stream>>>(u, 2 * BB * NP);
    gemm_qkT<<<4096, 256, 0, stream>>>(fq, fk, S, ST);

    for (int it = 0; it < 50; ++it) {
        lse_update<<<1024, 256, 0, stream>>>(S,  v, u);   // u half-step
        lse_update<<<1024, 256, 0, stream>>>(ST, u, v);   // v half-step
    }
    loss_kernel<<<1024, 256, 0, stream>>>(S, ST, u, v, ip, out);
}